// Topk_Router_8512625180881
// MI455X (gfx1250) — compile-verified
//
#include <hip/hip_runtime.h>
#include <math.h>

#define D_MODEL 2048
#define NUM_EXPERTS 64
#define TOKENS_PER_BLOCK 128
#define LDS_STRIDE 68   // 64 + 4 pad: lane-halves (token offset 8) land on disjoint banks

typedef __attribute__((ext_vector_type(2))) float v2f;
typedef __attribute__((ext_vector_type(8))) float v8f;

__global__ __launch_bounds__(256, 2)
void router_topk_kernel(const float* __restrict__ x,
                        const float* __restrict__ W,
                        const float* __restrict__ bias,
                        float* __restrict__ out_sf,
                        int* __restrict__ out_idx) {
    __shared__ float s_logits[TOKENS_PER_BLOCK * LDS_STRIDE];
    __shared__ float s_bias[NUM_EXPERTS];

    const int tid  = threadIdx.x;
    const int lane = tid & 31;
    const int wave = tid >> 5;

    if (tid < NUM_EXPERTS) s_bias[tid] = bias[tid];

    // ---- GEMM phase: wave computes 16 tokens x 64 experts, fp32 WMMA 16x16x4 ----
    // A (16x4): lanes 0-15 row=lane, K={k,k+1}; lanes 16-31 row=lane-16, K={k+2,k+3}
    // B (4x16): lanes 0-15 col=lane, K={k,k+1}; lanes 16-31 col=lane-16, K={k+2,k+3}
    const int rowBase = blockIdx.x * TOKENS_PER_BLOCK + wave * 16;
    const int m    = lane & 15;
    const int kOff = (lane >> 4) * 2;

    const float* xrow = x + (size_t)(rowBase + m) * D_MODEL + kOff;
    const float* w0   = W + (size_t)( 0 + m) * D_MODEL + kOff;
    const float* w1   = W + (size_t)(16 + m) * D_MODEL + kOff;
    const float* w2   = W + (size_t)(32 + m) * D_MODEL + kOff;
    const float* w3   = W + (size_t)(48 + m) * D_MODEL + kOff;

    v8f acc0 = {}, acc1 = {}, acc2 = {}, acc3 = {};

#pragma unroll 4
    for (int k = 0; k < D_MODEL; k += 4) {
        v2f a  = *(const v2f*)(xrow + k);
        v2f b0 = *(const v2f*)(w0 + k);
        v2f b1 = *(const v2f*)(w1 + k);
        v2f b2 = *(const v2f*)(w2 + k);
        v2f b3 = *(const v2f*)(w3 + k);
        // 8 args: (neg_a, A, neg_b, B, c_mod, C, reuse_a, reuse_b)
        acc0 = __builtin_amdgcn_wmma_f32_16x16x4_f32(false, a, false, b0, (short)0, acc0, false, false);
        acc1 = __builtin_amdgcn_wmma_f32_16x16x4_f32(false, a, false, b1, (short)0, acc1, false, false);
        acc2 = __builtin_amdgcn_wmma_f32_16x16x4_f32(false, a, false, b2, (short)0, acc2, false, false);
        acc3 = __builtin_amdgcn_wmma_f32_16x16x4_f32(false, a, false, b3, (short)0, acc3, false, false);
    }

    // ---- Spill logits to LDS in [token][expert] layout ----
    // C/D layout: VGPR r holds M=r (lanes 0-15) / M=r+8 (lanes 16-31), N = lane&15
    const int tbase = wave * 16 + (lane >> 4) * 8;
    const int e0    = lane & 15;
#pragma unroll
    for (int r = 0; r < 8; ++r) {
        s_logits[(tbase + r) * LDS_STRIDE + e0 +  0] = acc0[r];
        s_logits[(tbase + r) * LDS_STRIDE + e0 + 16] = acc1[r];
        s_logits[(tbase + r) * LDS_STRIDE + e0 + 32] = acc2[r];
        s_logits[(tbase + r) * LDS_STRIDE + e0 + 48] = acc3[r];
    }
    __syncthreads();

    // ---- Per-token top-2 + sparse softmax (1 thread per token) ----
    if (tid < TOKENS_PER_BLOCK) {
        const int gtok = blockIdx.x * TOKENS_PER_BLOCK + tid;
        const float* row = &s_logits[tid * LDS_STRIDE];

        float m1 = -INFINITY, m2 = -INFINITY;
        int   i1 = 0,         i2 = 0;
#pragma unroll
        for (int e = 0; e < NUM_EXPERTS; ++e) {
            float v = row[e] + s_bias[e];
            if (v > m1)      { m2 = m1; i2 = i1; m1 = v; i1 = e; }
            else if (v > m2) { m2 = v;  i2 = e; }
        }
        // softmax over {m1, m2} with -inf elsewhere
        float e2  = __expf(m2 - m1);
        float inv = 1.0f / (1.0f + e2);

        float*  orow = out_sf + (size_t)gtok * NUM_EXPERTS;
        float4  z    = make_float4(0.f, 0.f, 0.f, 0.f);
#pragma unroll
        for (int q = 0; q < NUM_EXPERTS / 4; ++q) ((float4*)orow)[q] = z;
        orow[i1] = inv;
        orow[i2] = e2 * inv;

        out_idx[gtok * 2 + 0] = i1;
        out_idx[gtok * 2 + 1] = i2;
    }
}

extern "C" void kernel_launch(void* const* d_in, const int* in_sizes, int n_in,
                              void* d_out, int out_size, void* d_ws, size_t ws_size,
                              hipStream_t stream) {
    const float* x = (const float*)d_in[0];   // [4,4096,2048]
    const float* W = (const float*)d_in[1];   // [64,2048]
    const float* b = (const float*)d_in[2];   // [64]

    const int n_tokens = in_sizes[0] / D_MODEL;           // 16384
    float* out_sf  = (float*)d_out;                       // [n_tokens,64] f32
    int*   out_idx = (int*)((float*)d_out + (size_t)n_tokens * NUM_EXPERTS); // [n_tokens,2] i32

    const int blocks = n_tokens / TOKENS_PER_BLOCK;       // 128
    router_topk_kernel<<<blocks, 256, 0, stream>>>(x, W, b, out_sf, out_idx);
}